// HRNet_81982335746521
// MI455X (gfx1250) — compile-verified
//
#include <hip/hip_runtime.h>
#include <stdint.h>

typedef float v2f __attribute__((ext_vector_type(2)));
typedef float v8f __attribute__((ext_vector_type(8)));

#define NVOX   150000
#define SORTN  262144      // 2^18 >= NVOX, padded bitonic size
#define SZd    41
#define SYd    1600
#define SXd    1408
#define NTILES (NVOX / 16) // 9375 exactly
#define WPB    8           // waves per block in conv kernels

#define LSORT_ELEMS   2048
#define LSORT_THREADS 1024

// ---------------------------------------------------------------- utilities
__global__ void k_zero(float* p, int n) {
    int i = blockIdx.x * blockDim.x + threadIdx.x;
    if (i < n) p[i] = 0.f;
}

// lin index + bitonic pad init
__global__ void k_lininit(const int* __restrict__ idx, uint32_t* __restrict__ key,
                          int* __restrict__ val) {
    int i = blockIdx.x * blockDim.x + threadIdx.x;
    if (i >= SORTN) return;
    if (i < NVOX) {
        int b = idx[i * 4 + 0], z = idx[i * 4 + 1], y = idx[i * 4 + 2], x = idx[i * 4 + 3];
        key[i] = (uint32_t)(((b * SZd + z) * SYd + y) * SXd + x);
        val[i] = i;
    } else {
        key[i] = 0x7FFFFFFFu;   // pad sorts to the end
        val[i] = -1;
    }
}

// global bitonic pass (only for span j >= LSORT_ELEMS)
__global__ void k_bitonic(uint32_t* __restrict__ key, int* __restrict__ val,
                          int j, int k) {
    int i = blockIdx.x * blockDim.x + threadIdx.x;
    int ixj = i ^ j;
    if (ixj > i) {
        uint32_t ki = key[i], kj = key[ixj];
        bool asc = ((i & k) == 0);
        if ((ki > kj) == asc) {
            key[i] = kj; key[ixj] = ki;
            int vi = val[i]; val[i] = val[ixj]; val[ixj] = vi;
        }
    }
}

// LDS-fused bitonic: runs stages kstart..kend with all spans j < LSORT_ELEMS
// resolved inside one kernel (directions use GLOBAL indices, so the network
// is identical to the pure-global version).
__global__ void __launch_bounds__(LSORT_THREADS)
k_bitonic_local(uint32_t* __restrict__ key, int* __restrict__ val,
                int kstart, int kend) {
    __shared__ uint32_t lk[LSORT_ELEMS];
    __shared__ int      lv[LSORT_ELEMS];
    int base = blockIdx.x * LSORT_ELEMS;
    for (int t = threadIdx.x; t < LSORT_ELEMS; t += LSORT_THREADS) {
        lk[t] = key[base + t]; lv[t] = val[base + t];
    }
    __syncthreads();
    for (int k = kstart; k <= kend; k <<= 1) {
        int jstart = (k <= LSORT_ELEMS) ? (k >> 1) : (LSORT_ELEMS >> 1);
        for (int j = jstart; j > 0; j >>= 1) {
            int t = threadIdx.x;
            int i = (t & (j - 1)) + ((t & ~(j - 1)) << 1);
            int ixj = i + j;
            bool asc = (((base + i) & k) == 0);
            uint32_t ki = lk[i], kj = lk[ixj];
            if ((ki > kj) == asc) {
                lk[i] = kj; lk[ixj] = ki;
                int v = lv[i]; lv[i] = lv[ixj]; lv[ixj] = v;
            }
            __syncthreads();
        }
    }
    for (int t = threadIdx.x; t < LSORT_ELEMS; t += LSORT_THREADS) {
        key[base + t] = lk[t]; val[base + t] = lv[t];
    }
}

// rank[orig_row] = sorted position (== rank of its linear index)
__global__ void k_rank(const int* __restrict__ sval, int* __restrict__ rank) {
    int s = blockIdx.x * blockDim.x + threadIdx.x;
    if (s >= SORTN) return;
    int v = sval[s];
    if (v >= 0) rank[v] = s;
}

// rulebook (transposed [27][NVOX] for coalesced conv reads):
// neighbor's original row or -1
__global__ void k_rulebook(const int* __restrict__ idx, const uint32_t* __restrict__ skey,
                           const int* __restrict__ sval, int* __restrict__ rbT) {
    int t = blockIdx.x * blockDim.x + threadIdx.x;
    if (t >= NVOX * 27) return;
    int n = t / 27, o = t - n * 27;
    int b = idx[n * 4 + 0];
    int z = idx[n * 4 + 1] + (o / 9) - 1;
    int y = idx[n * 4 + 2] + ((o / 3) % 3) - 1;
    int x = idx[n * 4 + 3] + (o % 3) - 1;
    bool valid = (z >= 0) & (z < SZd) & (y >= 0) & (y < SYd) & (x >= 0) & (x < SXd);
    uint32_t nlin = (uint32_t)(((b * SZd + z) * SYd + y) * SXd + x);
    // lower_bound over the N real sorted keys (matches searchsorted 'left' + clip)
    int lo = 0, hi = NVOX;
    while (lo < hi) { int mid = (lo + hi) >> 1; if (skey[mid] < nlin) lo = mid + 1; else hi = mid; }
    int pos = lo < NVOX ? lo : NVOX - 1;
    rbT[o * NVOX + n] = (valid && skey[pos] == nlin) ? sval[pos] : -1;
}

// ------------------------------------------------------- WMMA conv kernels
// One wave == one 16-voxel tile; Cout=16 == WMMA N dim; f32 16x16x4 WMMA.
// Branch-free gathers: unconditional load from clamped row + cndmask zeroing,
// so the scheduler can hoist gathers ahead of the consuming wmma.

__global__ void __launch_bounds__(32 * WPB)
k_conv_in(const float* __restrict__ feats, const int* __restrict__ rbT,
          const float* __restrict__ w, float* __restrict__ out) {
    __shared__ float ws[27 * 4 * 16];
    for (int t = threadIdx.x; t < 27 * 4 * 16; t += blockDim.x) ws[t] = w[t];
    __syncthreads();
    int lane = threadIdx.x & 31;
    int tile = blockIdx.x * WPB + (threadIdx.x >> 5);
    if (tile >= NTILES) return;                 // wave-uniform
    int m = lane & 15, hi = lane >> 4;
    int voxel = tile * 16 + m;
    v8f acc = {0.f, 0.f, 0.f, 0.f, 0.f, 0.f, 0.f, 0.f};
#pragma unroll 3
    for (int k = 0; k < 27; ++k) {
        int row  = rbT[k * NVOX + voxel];
        int srow = row < 0 ? 0 : row;
        const float* p = feats + srow * 4 + 2 * hi;
        float ax = p[0], ay = p[1];
        bool ok = row >= 0;
        v2f a; a.x = ok ? ax : 0.f; a.y = ok ? ay : 0.f;
        const float* bw = ws + (k * 4 + 2 * hi) * 16 + m;
        v2f bb; bb.x = bw[0]; bb.y = bw[16];
        acc = __builtin_amdgcn_wmma_f32_16x16x4_f32(false, a, false, bb,
                                                    (short)0, acc, false, false);
    }
    float* o = out + (tile * 16 + 8 * hi) * 16 + m;
#pragma unroll
    for (int r = 0; r < 8; ++r) o[r * 16] = acc[r];
}

// conv1 + conv2 fused: A fragments gathered once, two B/C chains
__global__ void __launch_bounds__(32 * WPB)
k_conv16_dual(const float* __restrict__ x, const int* __restrict__ rbT,
              const float* __restrict__ w1, const float* __restrict__ w2,
              float* __restrict__ out1, float* __restrict__ out2) {
    __shared__ float ws1[27 * 16 * 16];
    __shared__ float ws2[27 * 16 * 16];
    for (int t = threadIdx.x; t < 27 * 16 * 16; t += blockDim.x) {
        ws1[t] = w1[t]; ws2[t] = w2[t];
    }
    __syncthreads();
    int lane = threadIdx.x & 31;
    int tile = blockIdx.x * WPB + (threadIdx.x >> 5);
    if (tile >= NTILES) return;                 // wave-uniform
    int m = lane & 15, hi = lane >> 4;
    int voxel = tile * 16 + m;
    v8f acc1 = {0.f, 0.f, 0.f, 0.f, 0.f, 0.f, 0.f, 0.f};
    v8f acc2 = {0.f, 0.f, 0.f, 0.f, 0.f, 0.f, 0.f, 0.f};
#pragma unroll 3
    for (int k = 0; k < 27; ++k) {
        int row  = rbT[k * NVOX + voxel];
        int srow = row < 0 ? 0 : row;
        bool ok  = row >= 0;
        const float* arow = x + srow * 16 + 2 * hi;
#pragma unroll
        for (int cb = 0; cb < 16; cb += 4) {
            float ax = arow[cb], ay = arow[cb + 1];
            v2f a; a.x = ok ? ax : 0.f; a.y = ok ? ay : 0.f;
            int base = (k * 16 + cb + 2 * hi) * 16 + m;
            v2f b1; b1.x = ws1[base]; b1.y = ws1[base + 16];
            v2f b2; b2.x = ws2[base]; b2.y = ws2[base + 16];
            acc1 = __builtin_amdgcn_wmma_f32_16x16x4_f32(false, a, false, b1,
                                                         (short)0, acc1, false, false);
            acc2 = __builtin_amdgcn_wmma_f32_16x16x4_f32(false, a, false, b2,
                                                         (short)0, acc2, false, false);
        }
    }
    float* o1 = out1 + (tile * 16 + 8 * hi) * 16 + m;
    float* o2 = out2 + (tile * 16 + 8 * hi) * 16 + m;
#pragma unroll
    for (int r = 0; r < 8; ++r) { o1[r * 16] = acc1[r]; o2[r * 16] = acc2[r]; }
}

// ----------------------------------------------------------------- BN+ReLU
__global__ void __launch_bounds__(256)
k_stats(const float* __restrict__ x, float* __restrict__ st) {
    __shared__ float ls[256], lq[256];
    int c = threadIdx.x & 15;
    int g = threadIdx.x >> 4;
    float s = 0.f, q = 0.f;
    for (int r = blockIdx.x * 16 + g; r < NVOX; r += gridDim.x * 16) {
        float v = x[r * 16 + c];
        s += v; q += v * v;
    }
    ls[threadIdx.x] = s; lq[threadIdx.x] = q;
    __syncthreads();
    for (int k = 128; k >= 16; k >>= 1) {
        if (threadIdx.x < k) {
            ls[threadIdx.x] += ls[threadIdx.x + k];
            lq[threadIdx.x] += lq[threadIdx.x + k];
        }
        __syncthreads();
    }
    if (threadIdx.x < 16) {
        atomicAdd(&st[c], ls[threadIdx.x]);
        atomicAdd(&st[16 + c], lq[threadIdx.x]);
    }
}

__global__ void __launch_bounds__(256)
k_bnrelu(float* __restrict__ x, const float* __restrict__ st,
         const float* __restrict__ g, const float* __restrict__ b) {
    int i = blockIdx.x * blockDim.x + threadIdx.x;
    if (i >= NVOX * 16) return;
    int c = i & 15;
    float mean = st[c] * (1.f / NVOX);
    float var  = st[16 + c] * (1.f / NVOX) - mean * mean;
    float y = (x[i] - mean) * rsqrtf(var + 1e-3f) * g[c] + b[c];
    x[i] = y > 0.f ? y : 0.f;
}

// cat_tensors collapses to a rank-gather of xa duplicated into both output
// halves; BN+ReLU of branch-a is fused here (xa has no other consumer).
__global__ void __launch_bounds__(256)
k_cat_bn(const float* __restrict__ raw, const float* __restrict__ st,
         const float* __restrict__ g, const float* __restrict__ b,
         const int* __restrict__ rank, float* __restrict__ out) {
    int i = blockIdx.x * blockDim.x + threadIdx.x;
    if (i >= NVOX * 16) return;
    int k = i >> 4, c = i & 15;
    float mean = st[c] * (1.f / NVOX);
    float var  = st[16 + c] * (1.f / NVOX) - mean * mean;
    float v = raw[rank[k] * 16 + c];
    float y = (v - mean) * rsqrtf(var + 1e-3f) * g[c] + b[c];
    y = y > 0.f ? y : 0.f;
    out[i] = y;
    out[i + NVOX * 16] = y;
}

// ------------------------------------------------------------------ driver
extern "C" void kernel_launch(void* const* d_in, const int* in_sizes, int n_in,
                              void* d_out, int out_size, void* d_ws, size_t ws_size,
                              hipStream_t stream) {
    const float* feats = (const float*)d_in[0];
    const int*   idx   = (const int*)d_in[1];
    const float* w_in  = (const float*)d_in[2];
    const float* g0    = (const float*)d_in[3];
    const float* b0    = (const float*)d_in[4];
    const float* w1    = (const float*)d_in[5];
    const float* g1    = (const float*)d_in[6];
    const float* b1    = (const float*)d_in[7];
    const float* w2    = (const float*)d_in[8];
    const float* g2    = (const float*)d_in[9];
    const float* b2    = (const float*)d_in[10];
    float* out = (float*)d_out;

    // workspace carve-up (256B aligned)
    char* ws = (char*)d_ws;
    size_t off = 0;
    auto take = [&](size_t bytes) { char* p = ws + off; off = (off + bytes + 255) & ~size_t(255); return p; };
    float*    st   = (float*)take(3 * 32 * sizeof(float));        // stats for 3 BN layers
    uint32_t* keys = (uint32_t*)take(SORTN * sizeof(uint32_t));
    int*      vals = (int*)take(SORTN * sizeof(int));
    int*      rank = (int*)take(NVOX * sizeof(int));
    int*      rbT  = (int*)take(NVOX * 27 * sizeof(int));
    float*    buf0 = (float*)take(NVOX * 16 * sizeof(float));     // conv_in -> x (in-place BN)
    float*    buf1 = (float*)take(NVOX * 16 * sizeof(float));     // conv1 raw (BN fused into cat)
    float*    buf2 = (float*)take(NVOX * 16 * sizeof(float));     // conv2 raw
    (void)ws_size; (void)n_in; (void)in_sizes; (void)out_size;

    // reset BN accumulators every call (graph replay does not re-poison)
    k_zero<<<1, 128, 0, stream>>>(st, 96);

    // sort linear indices: LDS-fused bitonic (36 launches instead of 171)
    k_lininit<<<SORTN / 256, 256, 0, stream>>>(idx, keys, vals);
    k_bitonic_local<<<SORTN / LSORT_ELEMS, LSORT_THREADS, 0, stream>>>(keys, vals, 2, LSORT_ELEMS);
    for (int k = LSORT_ELEMS * 2; k <= SORTN; k <<= 1) {
        for (int j = k >> 1; j >= LSORT_ELEMS; j >>= 1)
            k_bitonic<<<SORTN / 256, 256, 0, stream>>>(keys, vals, j, k);
        k_bitonic_local<<<SORTN / LSORT_ELEMS, LSORT_THREADS, 0, stream>>>(keys, vals, k, k);
    }
    k_rank<<<SORTN / 256, 256, 0, stream>>>(vals, rank);

    // shared rulebook for all three convs (transposed layout)
    k_rulebook<<<(NVOX * 27 + 255) / 256, 256, 0, stream>>>(idx, keys, vals, rbT);

    const int convBlocks = (NTILES + WPB - 1) / WPB;
    const int elemBlocks = (NVOX * 16 + 255) / 256;

    // conv_input (4->16) + BN + ReLU
    k_conv_in<<<convBlocks, 32 * WPB, 0, stream>>>(feats, rbT, w_in, buf0);
    k_stats<<<256, 256, 0, stream>>>(buf0, st + 0);
    k_bnrelu<<<elemBlocks, 256, 0, stream>>>(buf0, st + 0, g0, b0);

    // conv1 & conv2 (16->16) fused; branch-a BN deferred into the cat gather
    k_conv16_dual<<<convBlocks, 32 * WPB, 0, stream>>>(buf0, rbT, w1, w2, buf1, buf2);
    k_stats<<<256, 256, 0, stream>>>(buf1, st + 32);
    // branch-b BN kept for fidelity (its features do not reach the output)
    k_stats<<<256, 256, 0, stream>>>(buf2, st + 64);
    k_bnrelu<<<elemBlocks, 256, 0, stream>>>(buf2, st + 64, g2, b2);

    // CatTensors == rank-gather of BN(xa) into both output halves
    k_cat_bn<<<elemBlocks, 256, 0, stream>>>(buf1, st + 32, g1, b1, rank, out);
}